// ModelManifoldGeodesicFlow_16939351016036
// MI455X (gfx1250) — compile-verified
//
#include <hip/hip_runtime.h>
#include <math.h>

// ---------------------------------------------------------------------------
// ModelManifoldGeodesicFlow, closed-form CDNA5 implementation.
//
// D=320 (input), H=512 (hidden), M=128 (output of f).
//
// Heavy math reduced analytically:
//   K  = W1 W1^T                     (512x512, k=320)   -> WMMA GEMM
//   S  = W2^T W2                     (512x512, k=128)   -> WMMA GEMM
//   B  = S  @ (diag(c) K)            (512x512, k=512)   -> WMMA GEMM (B row scale)
//   M1 = (B diag(c)) @ S             (512x512, k=512)   -> WMMA GEMM (A col scale)
//   ||dG||_F^2 = 2 * sum_{h,h'} g_h g_h' K_hh' (K_hh' M1_hh' + B_hh' B_h'h)
//   Gamma.v.v  = 1.5 * W1^T(g*w*r) - 0.5 * W1^T(c*p)   (cheap matvec chain)
//   a = -(Gamma.v.v) / ((||dG||+1e-6) * (||v||+1e-6))
//   out = [v, a - 0.1*deviation]
// ---------------------------------------------------------------------------

#define D_IN  320
#define H_DIM 512
#define M_DIM 128
#define HH    (H_DIM * H_DIM)

typedef __attribute__((ext_vector_type(2))) float v2f;
typedef __attribute__((ext_vector_type(8))) float v8f;

// ---------------------------------------------------------------------------
// x = x0 + t*(x1-x0) + 4t(1-t)*deviation ;  zero the frobenius accumulator
// ---------------------------------------------------------------------------
__global__ void k_prepare(const float* __restrict__ t_, const float* __restrict__ state,
                          const float* __restrict__ x0, const float* __restrict__ x1,
                          float* __restrict__ x, float* __restrict__ scal) {
    int i = blockIdx.x * blockDim.x + threadIdx.x;
    float t = t_[0];
    if (i < D_IN) {
        float window = 4.0f * t * (1.0f - t);
        x[i] = x0[i] + t * (x1[i] - x0[i]) + window * state[i];
    }
    if (i == 0) scal[0] = 0.0f;
}

// ---------------------------------------------------------------------------
// Per hidden unit h: u=W1x, w=W1v, c=1-tanh(u)^2, g=-2*tanh(u)*c,
//                    y1=c*w, y2=g*w^2
// ---------------------------------------------------------------------------
__global__ void k_w1_forward(const float* __restrict__ W1, const float* __restrict__ x,
                             const float* __restrict__ v, float* __restrict__ c,
                             float* __restrict__ g, float* __restrict__ y1,
                             float* __restrict__ y2, float* __restrict__ w) {
    int h = blockIdx.x * blockDim.x + threadIdx.x;
    if (h >= H_DIM) return;
    const float* row = W1 + h * D_IN;
    float u = 0.f, wv = 0.f;
#pragma unroll 4
    for (int d = 0; d < D_IN; ++d) {
        float a = row[d];
        u  = fmaf(a, x[d], u);
        wv = fmaf(a, v[d], wv);
    }
    float s  = tanhf(u);
    float cc = 1.f - s * s;
    float gg = -2.f * s * cc;
    c[h] = cc; g[h] = gg; w[h] = wv;
    y1[h] = cc * wv;
    y2[h] = gg * wv * wv;
}

// q = W2 y1 ; z = W2 y2
__global__ void k_w2_forward(const float* __restrict__ W2, const float* __restrict__ y1,
                             const float* __restrict__ y2, float* __restrict__ q,
                             float* __restrict__ z) {
    int m = blockIdx.x * blockDim.x + threadIdx.x;
    if (m >= M_DIM) return;
    const float* row = W2 + m * H_DIM;
    float qq = 0.f, zz = 0.f;
#pragma unroll 4
    for (int h = 0; h < H_DIM; ++h) {
        float a = row[h];
        qq = fmaf(a, y1[h], qq);
        zz = fmaf(a, y2[h], zz);
    }
    q[m] = qq; z[m] = zz;
}

// r = W2^T q ; p = W2^T z ; t1 = g*w*r ; t2 = c*p
__global__ void k_w2t(const float* __restrict__ W2, const float* __restrict__ q,
                      const float* __restrict__ z, const float* __restrict__ c,
                      const float* __restrict__ g, const float* __restrict__ w,
                      float* __restrict__ t1, float* __restrict__ t2) {
    int h = blockIdx.x * blockDim.x + threadIdx.x;
    if (h >= H_DIM) return;
    float rr = 0.f, pp = 0.f;
#pragma unroll 4
    for (int m = 0; m < M_DIM; ++m) {
        float a = W2[m * H_DIM + h];
        rr = fmaf(a, q[m], rr);
        pp = fmaf(a, z[m], pp);
    }
    t1[h] = g[h] * w[h] * rr;
    t2[h] = c[h] * pp;
}

// anum[d] = -(1.5 * (W1^T t1)[d] - 0.5 * (W1^T t2)[d])
__global__ void k_w1t(const float* __restrict__ W1, const float* __restrict__ t1,
                      const float* __restrict__ t2, float* __restrict__ anum) {
    int d = blockIdx.x * blockDim.x + threadIdx.x;
    if (d >= D_IN) return;
    float d1 = 0.f, d2 = 0.f;
#pragma unroll 4
    for (int h = 0; h < H_DIM; ++h) {
        float a = W1[h * D_IN + d];
        d1 = fmaf(a, t1[h], d1);
        d2 = fmaf(a, t2[h], d2);
    }
    anum[d] = -(1.5f * d1 - 0.5f * d2);
}

// scal[1] = ||v||
__global__ void k_vnorm(const float* __restrict__ v, float* __restrict__ scal) {
    __shared__ float red[256];
    int tid = threadIdx.x;
    float acc = 0.f;
    for (int i = tid; i < D_IN; i += 256) { float a = v[i]; acc = fmaf(a, a, acc); }
    red[tid] = acc; __syncthreads();
    for (int s = 128; s > 0; s >>= 1) {
        if (tid < s) red[tid] += red[tid + s];
        __syncthreads();
    }
    if (tid == 0) scal[1] = sqrtf(red[0]);
}

// ---------------------------------------------------------------------------
// 512x512 f32 GEMM on V_WMMA_F32_16X16X4_F32. One 16x16 tile per wave32,
// 8 waves/block, 128 blocks = 1024 tiles. K stepped by 4, unrolled x4;
// K is a compile-time constant (320/128/512, all % 16 == 0) so there is no
// remainder loop. TRANSA/TRANSB pick the addressing; SCALEA/SCALEB fold
// diag(c) into the operand load. Contiguous operands load as b64.
// Fragment layouts per ISA 7.12.2 (32-bit A 16x4, 32-bit C/D 16x16):
//   A: lanes 0-15 hold K pair {0,1}, lanes 16-31 hold {2,3} (VGPR0/1)
//   C/D: VGPR r -> rows r (lanes 0-15) / r+8 (lanes 16-31)
// ---------------------------------------------------------------------------
template <int K, bool TRANSA, bool TRANSB, bool SCALEA, bool SCALEB>
__global__ __launch_bounds__(256) void k_wmma_gemm512(
    const float* __restrict__ A, const float* __restrict__ B, float* __restrict__ C,
    int lda, int ldb,
    const float* __restrict__ aScale, const float* __restrict__ bScale) {
    const int lane = threadIdx.x & 31;
    const int wave = blockIdx.x * 8 + (threadIdx.x >> 5);
    const int tileI = wave >> 5;   // 32 tiles per row of the 512x512 output
    const int tileJ = wave & 31;
    const int half = lane >> 4;    // K pair select: lanes 0-15 -> K{0,1}, 16-31 -> K{2,3}
    const int l = lane & 15;
    const int rowA = tileI * 16 + l;
    const int colB = tileJ * 16 + l;

    v8f acc = {};
#pragma unroll 4
    for (int k0 = 0; k0 < K; k0 += 4) {
        const int ka0 = k0 + 2 * half;     // even -> 8B-aligned vector loads below
        v2f av, bv;
        if constexpr (!TRANSA) {
            av = *(const v2f*)(A + rowA * lda + ka0);          // contiguous pair
        } else {
            av[0] = A[ka0 * lda + rowA];
            av[1] = A[(ka0 + 1) * lda + rowA];
        }
        if constexpr (SCALEA) av *= *(const v2f*)(aScale + ka0);
        if constexpr (TRANSB) {
            bv = *(const v2f*)(B + colB * ldb + ka0);          // contiguous pair
        } else {
            bv[0] = B[ka0 * ldb + colB];
            bv[1] = B[(ka0 + 1) * ldb + colB];
            if (k0 + 8 < K)                                    // gfx1250 prefetch path
                __builtin_prefetch(B + (ka0 + 8) * ldb + colB, 0, 1);
        }
        if constexpr (SCALEB) bv *= *(const v2f*)(bScale + ka0);
        // 8 args: (neg_a, A, neg_b, B, c_mod, C, reuse_a, reuse_b)
        acc = __builtin_amdgcn_wmma_f32_16x16x4_f32(false, av, false, bv,
                                                    (short)0, acc, false, false);
    }
    const int col = tileJ * 16 + l;
#pragma unroll
    for (int r = 0; r < 8; ++r) {
        int row = tileI * 16 + r + 8 * half;  // C/D layout: VGPR r -> rows r / r+8
        C[row * H_DIM + col] = acc[r];
    }
}

// scal[0] += sum over (h,h') of g_h g_h' K * (K*M1 + B * B^T)
__global__ void k_frob_reduce(const float* __restrict__ Km, const float* __restrict__ Bm,
                              const float* __restrict__ M1, const float* __restrict__ g,
                              float* __restrict__ scal) {
    __shared__ float red[256];
    int tid = threadIdx.x;
    float acc = 0.f;
    for (int idx = blockIdx.x * 256 + tid; idx < HH; idx += gridDim.x * 256) {
        int h  = idx >> 9;     // idx / 512
        int h2 = idx & 511;    // idx % 512
        float k  = Km[idx];
        float b  = Bm[idx];
        float bt = Bm[h2 * H_DIM + h];
        acc = fmaf(g[h] * g[h2] * k, fmaf(M1[idx], k, b * bt), acc);
    }
    red[tid] = acc; __syncthreads();
    for (int s = 128; s > 0; s >>= 1) {
        if (tid < s) red[tid] += red[tid + s];
        __syncthreads();
    }
    if (tid == 0) atomicAdd(&scal[0], red[0]);
}

// out = [v, anum/((||dG||+1e-6)(||v||+1e-6)) - 0.1*deviation]
__global__ void k_finalize(const float* __restrict__ state, const float* __restrict__ anum,
                           const float* __restrict__ scal, float* __restrict__ out) {
    int i = blockIdx.x * blockDim.x + threadIdx.x;
    if (i >= 2 * D_IN) return;
    if (i < D_IN) {
        out[i] = state[D_IN + i];   // v
    } else {
        int d = i - D_IN;
        float N  = sqrtf(fmaxf(2.f * scal[0], 0.f));
        float vn = scal[1];
        float a = anum[d] / ((N + 1e-6f) * (vn + 1e-6f));
        out[i] = a - 0.1f * state[d];
    }
}

extern "C" void kernel_launch(void* const* d_in, const int* in_sizes, int n_in,
                              void* d_out, int out_size, void* d_ws, size_t ws_size,
                              hipStream_t stream) {
    const float* t  = (const float*)d_in[0];
    const float* st = (const float*)d_in[1];   // [deviation(320) | v(320)]
    const float* x0 = (const float*)d_in[2];
    const float* x1 = (const float*)d_in[3];
    const float* W1 = (const float*)d_in[4];   // (512,320) row-major
    const float* W2 = (const float*)d_in[5];   // (128,512) row-major
    const float* v  = st + D_IN;
    float* out = (float*)d_out;

    float* ws   = (float*)d_ws;
    float* c    = ws + 0;
    float* g    = ws + 512;
    float* w    = ws + 1024;
    float* y1   = ws + 1536;
    float* y2   = ws + 2048;
    float* q    = ws + 2560;   // 128
    float* z    = ws + 2688;   // 128
    float* t1   = ws + 2816;
    float* t2   = ws + 3328;
    float* anum = ws + 3840;   // 320
    float* x    = ws + 4160;   // 320
    float* scal = ws + 4480;   // [frob_sumsq, vnorm]
    float* Km   = ws + 8192;        // 512x512
    float* Sm   = Km + HH;
    float* Bm   = Sm + HH;
    float* M1   = Bm + HH;

    // Vector / small pipeline
    k_prepare   <<<2, 256, 0, stream>>>(t, st, x0, x1, x, scal);
    k_w1_forward<<<2, 256, 0, stream>>>(W1, x, v, c, g, y1, y2, w);
    k_w2_forward<<<1, 128, 0, stream>>>(W2, y1, y2, q, z);
    k_w2t       <<<2, 256, 0, stream>>>(W2, q, z, c, g, w, t1, t2);
    k_w1t       <<<2, 256, 0, stream>>>(W1, t1, t2, anum);
    k_vnorm     <<<1, 256, 0, stream>>>(v, scal);

    // GEMM chain on WMMA f32 pipes (all outputs 512x512)
    // K = W1 @ W1^T           (k=320): A normal lda=320, B^T ldb=320
    k_wmma_gemm512<D_IN, false, true, false, false>
        <<<128, 256, 0, stream>>>(W1, W1, Km, D_IN, D_IN, nullptr, nullptr);
    // S = W2^T @ W2           (k=128): A^T lda=512, B normal ldb=512
    k_wmma_gemm512<M_DIM, true, false, false, false>
        <<<128, 256, 0, stream>>>(W2, W2, Sm, H_DIM, H_DIM, nullptr, nullptr);
    // B = S @ (diag(c) K)     (k=512): row scale on B operand
    k_wmma_gemm512<H_DIM, false, false, false, true>
        <<<128, 256, 0, stream>>>(Sm, Km, Bm, H_DIM, H_DIM, nullptr, c);
    // M1 = (B diag(c)) @ S    (k=512): col scale on A operand
    k_wmma_gemm512<H_DIM, false, false, true, false>
        <<<128, 256, 0, stream>>>(Bm, Sm, M1, H_DIM, H_DIM, c, nullptr);

    // ||dG||_F^2 reduction and final assembly
    k_frob_reduce<<<64, 256, 0, stream>>>(Km, Bm, M1, g, scal);
    k_finalize   <<<3, 256, 0, stream>>>(st, anum, scal, out);
}